// GATLayer_87393994539133
// MI455X (gfx1250) — compile-verified
//
#include <hip/hip_runtime.h>
#include <hip/hip_bf16.h>
#include <math.h>

typedef __attribute__((ext_vector_type(2))) float v2f;
typedef __attribute__((ext_vector_type(8))) float v8f;

#define HIDDEN 128
#define HEADS 4
#define OUTC 32
#define NEG_SLOPE 0.2f
#define BN_EPS 1e-5f

// ---------- helpers ----------

__device__ __forceinline__ void edge_sd(const long long* __restrict__ ei, int E, int e,
                                        int& s, int& d) {
    if (e < E) { s = (int)ei[e]; d = (int)ei[(long long)E + e]; }
    else       { s = e - E; d = e - E; }           // self loop
}

// correct float atomic max via sign-split int/uint atomics (emax init'd to -inf)
__device__ __forceinline__ void atomicMaxFloat(float* addr, float value) {
    if (value >= 0.0f)
        atomicMax((int*)addr, __float_as_int(value));
    else
        atomicMin((unsigned int*)addr, __float_as_uint(value));
}

// ---------- kernels ----------

// zero agg / stats, -inf emax, zero denom
__global__ void k_init(float* __restrict__ agg, float* __restrict__ emax,
                       float* __restrict__ denom, float* __restrict__ stats,
                       int nAgg, int nNH) {
    int i = blockIdx.x * blockDim.x + threadIdx.x;
    if (i < nAgg) agg[i] = 0.0f;
    if (i < nNH)  { emax[i] = __int_as_float(0xff800000); denom[i] = 0.0f; }
    if (i < 512)  stats[i] = 0.0f;
}

// x = h @ W  via V_WMMA_F32_16X16X4_F32; one wave per 16x16 output tile.
// M = N nodes (multiple of 16), K = N-cols = 128.
__global__ void k_gemm_wmma(const float* __restrict__ h, const float* __restrict__ W,
                            float* __restrict__ x, int mTiles) {
    const int wave = (blockIdx.x * blockDim.x + threadIdx.x) >> 5;
    const int lane = threadIdx.x & 31;
    const int nTiles = HIDDEN / 16;                 // 8
    const int tM = wave / nTiles;
    const int tN = wave % nTiles;
    if (tM >= mTiles) return;                       // whole-wave uniform: EXEC stays all-ones

    const int half = lane >> 4;                     // 0: K pair {0,1}, 1: K pair {2,3}
    const int l16  = lane & 15;
    const float* __restrict__ hrow = h + (size_t)(tM * 16 + l16) * HIDDEN;

    v8f acc = {};
    #pragma unroll
    for (int kk = 0; kk < HIDDEN; kk += 4) {
        // A 16x4 fp32: lane<16 holds K=kk..kk+1 of row l16; lane>=16 holds K=kk+2..kk+3
        v2f a = { hrow[kk + half * 2 + 0], hrow[kk + half * 2 + 1] };
        // B 4x16 fp32: row striped across lanes within a VGPR (mirror of A layout)
        const float* __restrict__ wb = W + (size_t)(kk + half * 2) * HIDDEN + tN * 16 + l16;
        v2f b = { wb[0], wb[HIDDEN] };
        acc = __builtin_amdgcn_wmma_f32_16x16x4_f32(false, a, false, b,
                                                    (short)0, acc, false, false);
    }

    // C/D 16x16 f32: VGPR r, lane<16 -> (M=r, N=lane); lane>=16 -> (M=r+8, N=lane-16)
    const int col = tN * 16 + l16;
    const int rowbase = tM * 16 + half * 8;
    #pragma unroll
    for (int r = 0; r < 8; ++r)
        x[(size_t)(rowbase + r) * HIDDEN + col] = acc[r];
}

// per-(node,head) attention logits: a_src[n,h] = <x[n,h,:], att_src[h,:]>
__global__ void k_alpha(const float* __restrict__ x, const float* __restrict__ att_src,
                        const float* __restrict__ att_dst, float* __restrict__ asrc,
                        float* __restrict__ adst, int N) {
    int i = blockIdx.x * blockDim.x + threadIdx.x;
    if (i >= N * HEADS) return;
    int n = i >> 2, hh = i & 3;
    const float* __restrict__ xr = x + (size_t)n * HIDDEN + hh * OUTC;
    const float* __restrict__ as = att_src + hh * OUTC;
    const float* __restrict__ ad = att_dst + hh * OUTC;
    float s = 0.0f, d = 0.0f;
    #pragma unroll
    for (int c = 0; c < OUTC; ++c) { s += xr[c] * as[c]; d += xr[c] * ad[c]; }
    asrc[i] = s; adst[i] = d;
}

__device__ __forceinline__ float edge_logit(const float* __restrict__ asrc,
                                            const float* __restrict__ adst,
                                            int s, int d, int hh) {
    float v = asrc[s * HEADS + hh] + adst[d * HEADS + hh];
    return v > 0.0f ? v : NEG_SLOPE * v;
}

// segment max over incoming edges (atomic float max)
__global__ void k_emax(const long long* __restrict__ ei, const float* __restrict__ asrc,
                       const float* __restrict__ adst, float* __restrict__ emax,
                       int E, int N) {
    int i = blockIdx.x * blockDim.x + threadIdx.x;
    int Etot = E + N;
    if (i >= Etot * HEADS) return;
    int e = i >> 2, hh = i & 3;
    int s, d; edge_sd(ei, E, e, s, d);
    atomicMaxFloat(&emax[d * HEADS + hh], edge_logit(asrc, adst, s, d, hh));
}

// segment sum of exp(e - emax[dst])
__global__ void k_denom(const long long* __restrict__ ei, const float* __restrict__ asrc,
                        const float* __restrict__ adst, const float* __restrict__ emax,
                        float* __restrict__ denom, int E, int N) {
    int i = blockIdx.x * blockDim.x + threadIdx.x;
    int Etot = E + N;
    if (i >= Etot * HEADS) return;
    int e = i >> 2, hh = i & 3;
    int s, d; edge_sd(ei, E, e, s, d);
    float v = edge_logit(asrc, adst, s, d, hh);
    atomicAdd(&denom[d * HEADS + hh], expf(v - emax[d * HEADS + hh]));
}

// weighted message scatter: agg[dst,c] += exp(e - emax) * x[src,c]  (divide later)
__global__ void k_scatter(const long long* __restrict__ ei, const float* __restrict__ x,
                          const float* __restrict__ asrc, const float* __restrict__ adst,
                          const float* __restrict__ emax, float* __restrict__ agg,
                          int E) {
    int e = blockIdx.x;                 // one block per edge (incl. self loops)
    int c = threadIdx.x;                // 0..127 channel
    int hh = c >> 5;
    int s, d; edge_sd(ei, E, e, s, d);
    float v = edge_logit(asrc, adst, s, d, hh);
    float ee = expf(v - emax[d * HEADS + hh]);
    atomicAdd(&agg[(size_t)d * HIDDEN + c], ee * x[(size_t)s * HIDDEN + c]);
}

// per-node normalize + bias (in place), accumulate BN sum / sumsq per channel
__global__ void k_norm_stats(const float* __restrict__ denom, const float* __restrict__ bias,
                             float* __restrict__ agg, float* __restrict__ stats, int N) {
    int c = threadIdx.x;                // 0..127
    int hh = c >> 5;
    float b = bias[c];
    float lsum = 0.0f, lsq = 0.0f;
    for (int n = blockIdx.x; n < N; n += gridDim.x) {
        float o = agg[(size_t)n * HIDDEN + c] / (denom[n * HEADS + hh] + 1e-16f) + b;
        agg[(size_t)n * HIDDEN + c] = o;
        lsum += o; lsq += o * o;
    }
    atomicAdd(&stats[c], lsum);
    atomicAdd(&stats[HIDDEN + c], lsq);
}

// finalize mean / rsqrt(var + eps)
__global__ void k_finstats(float* __restrict__ stats, int N) {
    int c = threadIdx.x;
    float invN = 1.0f / (float)N;
    float mean = stats[c] * invN;
    float var  = stats[HIDDEN + c] * invN - mean * mean;
    stats[2 * HIDDEN + c] = mean;
    stats[3 * HIDDEN + c] = rsqrtf(var + BN_EPS);
}

// BN affine -> ELU -> residual
__global__ void k_out(const float* __restrict__ agg, const float* __restrict__ stats,
                      const float* __restrict__ gamma, const float* __restrict__ beta,
                      const float* __restrict__ h, float* __restrict__ out, int total) {
    int i = blockIdx.x * blockDim.x + threadIdx.x;
    if (i >= total) return;
    int c = i & (HIDDEN - 1);
    float o = (agg[i] - stats[2 * HIDDEN + c]) * stats[3 * HIDDEN + c] * gamma[c] + beta[c];
    o = o > 0.0f ? o : expm1f(o);
    out[i] = o + h[i];
}

// ---------- launch ----------

extern "C" void kernel_launch(void* const* d_in, const int* in_sizes, int n_in,
                              void* d_out, int out_size, void* d_ws, size_t ws_size,
                              hipStream_t stream) {
    const float*     h       = (const float*)d_in[0];
    const long long* ei      = (const long long*)d_in[1];   // int64 [2, E]
    const float*     W       = (const float*)d_in[2];
    const float*     att_src = (const float*)d_in[3];
    const float*     att_dst = (const float*)d_in[4];
    const float*     bias    = (const float*)d_in[5];
    const float*     gamma   = (const float*)d_in[6];
    const float*     beta    = (const float*)d_in[7];
    float*           out     = (float*)d_out;

    const int N    = in_sizes[0] / HIDDEN;    // 50000
    const int E    = in_sizes[1] / 2;         // 800000
    const int Etot = E + N;                   // + self loops
    const int nAgg = N * HIDDEN;
    const int nNH  = N * HEADS;

    // workspace layout (floats)
    float* ws    = (float*)d_ws;
    float* x     = ws;                        // [N,128]
    float* agg   = x     + (size_t)nAgg;      // [N,128]
    float* emax  = agg   + (size_t)nAgg;      // [N,4]
    float* denom = emax  + (size_t)nNH;       // [N,4]
    float* stats = denom + (size_t)nNH;       // 512: sum|sumsq|mean|inv
    float* asrc  = stats + 512;               // [N,4]
    float* adst  = asrc  + (size_t)nNH;       // [N,4]

    // 1) init scratch
    k_init<<<(nAgg + 255) / 256, 256, 0, stream>>>(agg, emax, denom, stats, nAgg, nNH);

    // 2) x = h @ W via fp32 WMMA (N multiple of 16 -> exact tiling, EXEC all-ones)
    {
        const int mTiles = N / 16;
        const int waves  = mTiles * (HIDDEN / 16);
        k_gemm_wmma<<<(waves + 3) / 4, 128, 0, stream>>>(h, W, x, mTiles);
    }

    // 3) per-(node,head) logit tables
    k_alpha<<<(nNH + 255) / 256, 256, 0, stream>>>(x, att_src, att_dst, asrc, adst, N);

    // 4) segment max
    k_emax<<<(Etot * HEADS + 255) / 256, 256, 0, stream>>>(ei, asrc, adst, emax, E, N);

    // 5) segment sum of exp
    k_denom<<<(Etot * HEADS + 255) / 256, 256, 0, stream>>>(ei, asrc, adst, emax, denom, E, N);

    // 6) weighted scatter-add (one block per edge, one thread per channel)
    k_scatter<<<Etot, HIDDEN, 0, stream>>>(ei, x, asrc, adst, emax, agg, E);

    // 7) normalize + bias + BN partial sums
    k_norm_stats<<<256, HIDDEN, 0, stream>>>(denom, bias, agg, stats, N);

    // 8) BN stats finalize
    k_finstats<<<1, HIDDEN, 0, stream>>>(stats, N);

    // 9) BN affine + ELU + residual
    k_out<<<(nAgg + 255) / 256, 256, 0, stream>>>(agg, stats, gamma, beta, h, out, nAgg);
}